// MambaModule_55929064129097
// MI455X (gfx1250) — compile-verified
//
#include <hip/hip_runtime.h>
#include <hip/hip_bf16.h>
#include <math.h>

// ---------------- model dims ----------------
#define BB     4
#define LL     1024
#define DMODEL 512
#define DINNER 1024
#define NSTATE 16
#define DTRANK 32
#define XDBDIM 64      // DT_RANK + 2*D_STATE
#define DCONV  4
#define ROWS   (BB*LL) // 4096

typedef __attribute__((ext_vector_type(2))) float v2f;
typedef __attribute__((ext_vector_type(4))) float v4f;
typedef __attribute__((ext_vector_type(8))) float v8f;

#if __has_builtin(__builtin_amdgcn_sched_group_barrier)
#define SCHED_GROUP(mask, size, id) __builtin_amdgcn_sched_group_barrier(mask, size, id)
#else
#define SCHED_GROUP(mask, size, id)
#endif

__device__ __forceinline__ float silu_f(float x) {
    return x / (1.0f + __expf(-x));
}
__device__ __forceinline__ float softplus_f(float x) {
    return x > 20.0f ? x : log1pf(__expf(x));
}

// ---------------- embedding: h = x @ emb_W^T + b  (K=2) ----------------
__global__ void k_embed(const float* __restrict__ x, const float* __restrict__ W,
                        const float* __restrict__ b, float* __restrict__ h) {
    int i = blockIdx.x * blockDim.x + threadIdx.x;       // over ROWS*DMODEL
    if (i >= ROWS * DMODEL) return;
    int r = i / DMODEL, d = i % DMODEL;
    h[i] = x[r * 2 + 0] * W[d * 2 + 0] + x[r * 2 + 1] * W[d * 2 + 1] + b[d];
}

// ---------------- RMSNorm: one wave (32 lanes) per row ----------------
__global__ void k_rmsnorm(const float* __restrict__ in, const float* __restrict__ w,
                          float* __restrict__ out, int D) {
    int lane = threadIdx.x & 31;
    int wv   = threadIdx.x >> 5;
    int row  = blockIdx.x * (blockDim.x >> 5) + wv;      // grid sized exactly
    const float* p = in + (size_t)row * D;
    float s = 0.0f;
    for (int j = lane; j < D; j += 32) { float v = p[j]; s += v * v; }
    #pragma unroll
    for (int off = 16; off > 0; off >>= 1) s += __shfl_xor(s, off, 32);
    float inv = rsqrtf(s / (float)D + 1e-5f);
    float* o = out + (size_t)row * D;
    for (int j = lane; j < D; j += 32) o[j] = p[j] * inv * w[j];
}

// ---------------- fp32 WMMA GEMM: C = A @ W^T (+resid) ----------------
// A: (M,K) ld=lda ; W: (N,K) ld=ldw ; C: (M,N) ld=ldc
// One wave computes a 16x64 strip (1 A frag x 4 B frags, 4 accumulators).
//
// K is consumed in a permuted order (legal: reduction is order-free, A and B
// use the same permutation): half-wave hi sweeps physical K in
// [hi*K/2, (hi+1)*K/2), pairs {2s, 2s+1} per step s. Per lane this makes the
// fragment stream CONTIGUOUS, so one b128 load feeds two k-steps. Explicit
// double-buffering + sched_group_barrier pin a 1-load : 2-WMMA issue pattern
// so in-flight b128 loads overlap the previous iteration's WMMAs.
// Requires N % 64 == 0, M % 16 == 0, K % 8 == 0 (true for all uses here).
__global__ void k_gemm_wmma(const float* __restrict__ A, int lda,
                            const float* __restrict__ W, int ldw,
                            float* __restrict__ C, int ldc,
                            const float* __restrict__ resid,
                            int M, int N, int K) {
    int lane  = threadIdx.x & 31;
    int wv    = threadIdx.x >> 5;
    int strip = blockIdx.x * (blockDim.x >> 5) + wv;
    int stripsN = N >> 6;                                // N/64
    int totalStrips = (M >> 4) * stripsN;
    if (strip >= totalStrips) return;                    // wave-uniform
    int tm = strip / stripsN;
    int tn = strip - tm * stripsN;                       // 64-col group

    int hi = lane >> 4;        // half-wave: which half of K this lane sweeps
    int lo = lane & 15;        // M index for A frag, N index for B frags
    int half = K >> 1;

    const float* arow  = A + (size_t)(tm * 16 + lo) * lda + hi * half;
    const float* wrow0 = W + (size_t)(tn * 64 +  0 + lo) * ldw + hi * half;
    const float* wrow1 = W + (size_t)(tn * 64 + 16 + lo) * ldw + hi * half;
    const float* wrow2 = W + (size_t)(tn * 64 + 32 + lo) * ldw + hi * half;
    const float* wrow3 = W + (size_t)(tn * 64 + 48 + lo) * ldw + hi * half;

    v8f acc0 = {}, acc1 = {}, acc2 = {}, acc3 = {};

    int iters = K >> 3;                                  // 2 k-steps per iter
    // prologue: load iteration 0 fragments (b128 each = 2 k-steps)
    v4f a  = *(const v4f*)(arow);
    v4f b0 = *(const v4f*)(wrow0);
    v4f b1 = *(const v4f*)(wrow1);
    v4f b2 = *(const v4f*)(wrow2);
    v4f b3 = *(const v4f*)(wrow3);

    for (int it = 0; it < iters - 1; ++it) {
        // prefetch next iteration into distinct registers
        int o = (it + 1) << 2;
        v4f an  = *(const v4f*)(arow  + o);
        v4f c0  = *(const v4f*)(wrow0 + o);
        v4f c1  = *(const v4f*)(wrow1 + o);
        v4f c2  = *(const v4f*)(wrow2 + o);
        v4f c3  = *(const v4f*)(wrow3 + o);
        // 2 virtual k-steps x 4 accumulators = 8 WMMAs on current regs
        #pragma unroll
        for (int j = 0; j < 2; ++j) {
            v2f af, f0, f1, f2, f3;
            af.x = a [2 * j]; af.y = a [2 * j + 1];
            f0.x = b0[2 * j]; f0.y = b0[2 * j + 1];
            f1.x = b1[2 * j]; f1.y = b1[2 * j + 1];
            f2.x = b2[2 * j]; f2.y = b2[2 * j + 1];
            f3.x = b3[2 * j]; f3.y = b3[2 * j + 1];
            acc0 = __builtin_amdgcn_wmma_f32_16x16x4_f32(false, af, false, f0,
                                                         (short)0, acc0, false, false);
            acc1 = __builtin_amdgcn_wmma_f32_16x16x4_f32(false, af, false, f1,
                                                         (short)0, acc1, false, false);
            acc2 = __builtin_amdgcn_wmma_f32_16x16x4_f32(false, af, false, f2,
                                                         (short)0, acc2, false, false);
            acc3 = __builtin_amdgcn_wmma_f32_16x16x4_f32(false, af, false, f3,
                                                         (short)0, acc3, false, false);
        }
        a = an; b0 = c0; b1 = c1; b2 = c2; b3 = c3;
        // pin issue pattern: interleave the 5 VMEM reads with the 8 WMMAs
        SCHED_GROUP(0x020, 1, 0);   // VMEM read
        SCHED_GROUP(0x008, 2, 0);   // 2x WMMA
        SCHED_GROUP(0x020, 1, 0);
        SCHED_GROUP(0x008, 2, 0);
        SCHED_GROUP(0x020, 1, 0);
        SCHED_GROUP(0x008, 2, 0);
        SCHED_GROUP(0x020, 1, 0);
        SCHED_GROUP(0x008, 2, 0);
        SCHED_GROUP(0x020, 1, 0);
    }
    // epilogue: last iteration's WMMAs
    #pragma unroll
    for (int j = 0; j < 2; ++j) {
        v2f af, f0, f1, f2, f3;
        af.x = a [2 * j]; af.y = a [2 * j + 1];
        f0.x = b0[2 * j]; f0.y = b0[2 * j + 1];
        f1.x = b1[2 * j]; f1.y = b1[2 * j + 1];
        f2.x = b2[2 * j]; f2.y = b2[2 * j + 1];
        f3.x = b3[2 * j]; f3.y = b3[2 * j + 1];
        acc0 = __builtin_amdgcn_wmma_f32_16x16x4_f32(false, af, false, f0,
                                                     (short)0, acc0, false, false);
        acc1 = __builtin_amdgcn_wmma_f32_16x16x4_f32(false, af, false, f1,
                                                     (short)0, acc1, false, false);
        acc2 = __builtin_amdgcn_wmma_f32_16x16x4_f32(false, af, false, f2,
                                                     (short)0, acc2, false, false);
        acc3 = __builtin_amdgcn_wmma_f32_16x16x4_f32(false, af, false, f3,
                                                     (short)0, acc3, false, false);
    }

    // C/D layout: VGPR v -> row (v + 8*hi), col = lo (per 16-col tile)
    #pragma unroll
    for (int v = 0; v < 8; ++v) {
        int r = tm * 16 + v + 8 * hi;
        size_t base = (size_t)r * ldc + tn * 64 + lo;
        float v0 = acc0[v], v1 = acc1[v], v2 = acc2[v], v3 = acc3[v];
        if (resid) {
            v0 += resid[base +  0];
            v1 += resid[base + 16];
            v2 += resid[base + 32];
            v3 += resid[base + 48];
        }
        C[base +  0] = v0;
        C[base + 16] = v1;
        C[base + 32] = v2;
        C[base + 48] = v3;
    }
}

// ---------------- causal depthwise conv (D_CONV=4) + SiLU ----------------
// input = xz[:, :DINNER] (ld 2*DINNER), output u (ld DINNER)
__global__ void k_conv_silu(const float* __restrict__ xz, const float* __restrict__ cw,
                            const float* __restrict__ cb, float* __restrict__ u) {
    int i = blockIdx.x * blockDim.x + threadIdx.x;       // over ROWS*DINNER
    if (i >= ROWS * DINNER) return;
    int r  = i >> 10;            // row = b*L + l
    int ed = i & (DINNER - 1);
    int b  = r >> 10;
    int l  = r & (LL - 1);
    float acc = cb[ed];
    #pragma unroll
    for (int k = 0; k < DCONV; ++k) {
        int ls = l - (DCONV - 1) + k;
        if (ls >= 0)
            acc += xz[(size_t)(b * LL + ls) * (2 * DINNER) + ed] * cw[ed * DCONV + k];
    }
    u[i] = silu_f(acc);
}

// ---------------- selective scan: 1 thread per (b, ed, n) ----------------
// 16 consecutive lanes = one (b,ed) channel; y reduced via shfl_xor.
__global__ void k_scan(const float* __restrict__ u, const float* __restrict__ draw,
                       const float* __restrict__ xdb, const float* __restrict__ dtb,
                       const float* __restrict__ Alog, const float* __restrict__ Dskip,
                       float* __restrict__ y) {
    int gid = blockIdx.x * blockDim.x + threadIdx.x;     // 0 .. B*DINNER*NSTATE-1
    int n  = gid & (NSTATE - 1);
    int ch = gid >> 4;                                   // b*DINNER + ed
    int ed = ch & (DINNER - 1);
    int b  = ch >> 10;

    float Aa   = -__expf(Alog[ed * NSTATE + n]);
    float dtbe = dtb[ed];
    float Dk   = Dskip[ed];
    float h = 0.0f;

    for (int t = 0; t < LL; ++t) {
        size_t row = (size_t)(b * LL + t);
        float dr    = draw[row * DINNER + ed] + dtbe;
        float delta = softplus_f(dr);
        float uu    = u[row * DINNER + ed];
        float Bv    = xdb[row * XDBDIM + DTRANK + n];
        float Cv    = xdb[row * XDBDIM + DTRANK + NSTATE + n];
        h = __expf(delta * Aa) * h + (delta * Bv) * uu;
        float p = h * Cv;
        p += __shfl_xor(p, 1, 32);
        p += __shfl_xor(p, 2, 32);
        p += __shfl_xor(p, 4, 32);
        p += __shfl_xor(p, 8, 32);
        if (n == 0) y[row * DINNER + ed] = p + Dk * uu;
    }
}

// ---------------- output gate: y *= silu(z), z = xz[:, DINNER:] ----------------
__global__ void k_gate(float* __restrict__ y, const float* __restrict__ xz) {
    int i = blockIdx.x * blockDim.x + threadIdx.x;       // over ROWS*DINNER
    if (i >= ROWS * DINNER) return;
    int r  = i >> 10;
    int ed = i & (DINNER - 1);
    float z = xz[(size_t)r * (2 * DINNER) + DINNER + ed];
    y[i] *= silu_f(z);
}

extern "C" void kernel_launch(void* const* d_in, const int* in_sizes, int n_in,
                              void* d_out, int out_size, void* d_ws, size_t ws_size,
                              hipStream_t stream) {
    const float* x       = (const float*)d_in[0];
    const float* emb_W   = (const float*)d_in[1];
    const float* emb_b   = (const float*)d_in[2];
    const float* norm_w  = (const float*)d_in[3];
    const float* in_proj = (const float*)d_in[4];
    const float* conv_W  = (const float*)d_in[5];
    const float* conv_b  = (const float*)d_in[6];
    const float* x_proj  = (const float*)d_in[7];
    const float* dt_proj = (const float*)d_in[8];
    const float* dt_b    = (const float*)d_in[9];
    const float* A_log   = (const float*)d_in[10];
    const float* Dskip   = (const float*)d_in[11];
    const float* out_prj = (const float*)d_in[12];
    const float* normf_w = (const float*)d_in[13];

    // workspace layout (floats)
    float* ws   = (float*)d_ws;
    float* hbuf = ws;                                    // 4096*512
    float* xz   = hbuf + (size_t)ROWS * DMODEL;          // 4096*2048
    float* ubuf = xz   + (size_t)ROWS * 2 * DINNER;      // 4096*1024
    float* xdb  = ubuf + (size_t)ROWS * DINNER;          // 4096*64
    float* ybuf = xdb  + (size_t)ROWS * XDBDIM;          // 4096*1024
    float* tmp  = ybuf + (size_t)ROWS * DINNER;          // 4096*1024 (hn aliases draw)
    float* hn   = tmp;
    float* draw = tmp;

    const int T = 256;

    k_embed<<<(ROWS * DMODEL + T - 1) / T, T, 0, stream>>>(x, emb_W, emb_b, hbuf);

    for (int l = 0; l < 2; ++l) {
        const float* nw  = norm_w  + (size_t)l * DMODEL;
        const float* ipW = in_proj + (size_t)l * 2 * DINNER * DMODEL;
        const float* cW  = conv_W  + (size_t)l * DINNER * DCONV;
        const float* cb  = conv_b  + (size_t)l * DINNER;
        const float* xpW = x_proj  + (size_t)l * XDBDIM * DINNER;
        const float* dtW = dt_proj + (size_t)l * DINNER * DTRANK;
        const float* dtb = dt_b    + (size_t)l * DINNER;
        const float* Al  = A_log   + (size_t)l * DINNER * NSTATE;
        const float* Dk  = Dskip   + (size_t)l * DINNER;
        const float* opW = out_prj + (size_t)l * DMODEL * DINNER;

        // hn = rmsnorm(h)
        k_rmsnorm<<<ROWS / 8, T, 0, stream>>>(hbuf, nw, hn, DMODEL);

        // xz = hn @ ipW^T : M=4096 N=2048 K=512 -> (256*32) strips / 8 waves
        k_gemm_wmma<<<(ROWS / 16) * (2 * DINNER / 64) / 8, T, 0, stream>>>(
            hn, DMODEL, ipW, DMODEL, xz, 2 * DINNER, nullptr,
            ROWS, 2 * DINNER, DMODEL);

        // u = silu(causal_conv(xz[:, :DINNER]))
        k_conv_silu<<<(ROWS * DINNER + T - 1) / T, T, 0, stream>>>(xz, cW, cb, ubuf);

        // xdb = u @ xpW^T : M=4096 N=64 K=1024 -> (256*1) strips / 8 waves
        k_gemm_wmma<<<(ROWS / 16) * (XDBDIM / 64) / 8, T, 0, stream>>>(
            ubuf, DINNER, xpW, DINNER, xdb, XDBDIM, nullptr,
            ROWS, XDBDIM, DINNER);

        // draw = xdb[:, :32] @ dtW^T : M=4096 N=1024 K=32 (softplus+bias in scan)
        k_gemm_wmma<<<(ROWS / 16) * (DINNER / 64) / 8, T, 0, stream>>>(
            xdb, XDBDIM, dtW, DTRANK, draw, DINNER, nullptr,
            ROWS, DINNER, DTRANK);

        // selective scan: 65536 threads
        k_scan<<<(BB * DINNER * NSTATE) / T, T, 0, stream>>>(
            ubuf, draw, xdb, dtb, Al, Dk, ybuf);

        // gate
        k_gate<<<(ROWS * DINNER + T - 1) / T, T, 0, stream>>>(ybuf, xz);

        // h += y @ opW^T : M=4096 N=512 K=1024, residual fused
        k_gemm_wmma<<<(ROWS / 16) * (DMODEL / 64) / 8, T, 0, stream>>>(
            ybuf, DINNER, opW, DINNER, hbuf, DMODEL, hbuf,
            ROWS, DMODEL, DINNER);
    }

    // final rmsnorm -> d_out
    k_rmsnorm<<<ROWS / 8, T, 0, stream>>>(hbuf, normf_w, (float*)d_out, DMODEL);
}